// GatedMDTA_42176578847012
// MI455X (gfx1250) — compile-verified
//
#include <hip/hip_runtime.h>
#include <math.h>

// Problem constants (match reference)
#define BB      2
#define CC      192
#define HH      256
#define WWDIM   256
#define HWN     (HH * WWDIM)     // 65536
#define QCC     (4 * CC)         // 768
#define NHEADS  6
#define HD      32               // CC / NHEADS
#define KSLICES 64
#define EPSF    1e-12f

typedef __attribute__((ext_vector_type(2))) float v2f;
typedef __attribute__((ext_vector_type(8))) float v8f;

// fp32 WMMA: D(16x16) += A(16x4) * B(4x16), full precision (matches fp32 reference)
__device__ __forceinline__ v8f wmma4(v2f a, v2f b, v8f c) {
  return __builtin_amdgcn_wmma_f32_16x16x4_f32(false, a, false, b, (short)0, c,
                                               false, false);
}

// ---------------------------------------------------------------------------
// Kernels 1/7: O[b][m][n] = sum_k Wt[m][k] * X[b][k][n] + bias[m]
// K = CC (192), Npix = HWN (65536) fixed so row strides fold into the signed
// 24-bit instruction offsets; inner loop: 2x v_add_nc_u32 + 9 loads + 4 WMMA.
// grid: (HWN/64, M/64, B), block 128 (4 waves); wave -> 16x64 output strip.
// ---------------------------------------------------------------------------
__global__ __launch_bounds__(128) void gemm_wmma_kernel(
    const float* __restrict__ Wt, const float* __restrict__ X,
    const float* __restrict__ bias, float* __restrict__ O)
{
  const int lane = threadIdx.x & 31;
  const int wave = threadIdx.x >> 5;
  const int l  = lane & 15;
  const int lh = lane >> 4;                 // 0: K={k,k+1}, 1: K={k+2,k+3}
  const int m0 = blockIdx.y * 64 + wave * 16;
  const int n0 = blockIdx.x * 64;

  const float* Xb = X + (size_t)blockIdx.z * CC * HWN;  // uniform SGPR base
  unsigned aoff = (unsigned)(m0 + l) * CC + 2u * (unsigned)lh;
  unsigned boff = (unsigned)(2 * lh) * HWN + (unsigned)(n0 + l);

  v8f acc0 = {}, acc1 = {}, acc2 = {}, acc3 = {};
  for (int k = 0; k < CC; k += 4) {
    v2f a = *(const v2f*)(Wt + aoff);                   // b64, 8B aligned
    v2f b0, b1, b2, b3;
    b0.x = Xb[boff];       b0.y = Xb[boff + HWN];       // +HWN folds into imm
    b1.x = Xb[boff + 16];  b1.y = Xb[boff + HWN + 16];
    b2.x = Xb[boff + 32];  b2.y = Xb[boff + HWN + 32];
    b3.x = Xb[boff + 48];  b3.y = Xb[boff + HWN + 48];
    acc0 = wmma4(a, b0, acc0);
    acc1 = wmma4(a, b1, acc1);
    acc2 = wmma4(a, b2, acc2);
    acc3 = wmma4(a, b3, acc3);
    aoff += 4u;
    boff += 4u * HWN;
  }

  float* Ob = O + (size_t)blockIdx.z * (unsigned)(gridDim.y * 64) * HWN;
  unsigned ooff = (unsigned)(m0 + 8 * lh) * HWN + (unsigned)(n0 + l);
  unsigned bix  = (unsigned)(m0 + 8 * lh);
#pragma unroll
  for (int r = 0; r < 8; ++r) {
    float bv = bias[bix + r];
    float* orow = Ob + ooff + (unsigned)r * HWN;        // r*HWN folds into imm
    orow[0]  = acc0[r] + bv;
    orow[16] = acc1[r] + bv;
    orow[32] = acc2[r] + bv;
    orow[48] = acc3[r] + bv;
  }
}

// ---------------------------------------------------------------------------
// Kernel 2: depthwise 3x3 conv (pad 1) + bias, NCHW, groups = QCC
// ---------------------------------------------------------------------------
__global__ __launch_bounds__(256) void dwconv_kernel(
    const float* __restrict__ in, const float* __restrict__ w9,
    const float* __restrict__ bias, float* __restrict__ out)
{
  size_t idx = (size_t)blockIdx.x * blockDim.x + threadIdx.x;
  const size_t total = (size_t)BB * QCC * HWN;
  if (idx >= total) return;
  int x  = (int)(idx % WWDIM);
  int y  = (int)((idx / WWDIM) % HH);
  int ch = (int)((idx / HWN) % QCC);
  int b  = (int)(idx / ((size_t)QCC * HWN));

  const float* wp   = w9 + ch * 9;
  const float* base = in + ((size_t)b * QCC + ch) * HWN;
  float s = bias[ch];
#pragma unroll
  for (int dy = -1; dy <= 1; ++dy) {
    int yy = y + dy;
    if (yy < 0 || yy >= HH) continue;
#pragma unroll
    for (int dx = -1; dx <= 1; ++dx) {
      int xx = x + dx;
      if (xx < 0 || xx >= WWDIM) continue;
      s += wp[(dy + 1) * 3 + (dx + 1)] * base[yy * WWDIM + xx];
    }
  }
  out[idx] = s;
}

// ---------------------------------------------------------------------------
// Kernel 3: inverse L2 norms of q rows (channels [0,C)) and k rows ([2C,3C)).
// rows = 2*B*C; one block per row.
// ---------------------------------------------------------------------------
__global__ __launch_bounds__(256) void rownorm_kernel(
    const float* __restrict__ qkv, float* __restrict__ invn)
{
  __shared__ float red[256];
  int row   = blockIdx.x;
  int which = row / (BB * CC);           // 0 = q, 1 = k
  int rem   = row % (BB * CC);
  int b = rem / CC;
  int c = rem % CC;
  int ch = (which == 0) ? c : (2 * CC + c);
  const float* p = qkv + ((size_t)b * QCC + ch) * HWN;

  float s = 0.f;
  for (int i = threadIdx.x; i < HWN; i += 256) { float v = p[i]; s += v * v; }
  red[threadIdx.x] = s;
  __syncthreads();
  for (int st = 128; st > 0; st >>= 1) {
    if ((int)threadIdx.x < st) red[threadIdx.x] += red[threadIdx.x + st];
    __syncthreads();
  }
  if (threadIdx.x == 0) invn[row] = 1.0f / fmaxf(sqrtf(red[0]), EPSF);
}

// ---------------------------------------------------------------------------
// Kernel 4: split-K raw q.k^T per head via WMMA. K = n (contiguous dim).
// q and k fragments share the same numeric offset; the second M/N tile is
// +16*HWN elements (4MB) which folds into the 24-bit immediate.
// grid: (KSLICES/4, B*HEADS), block 128; wave = one k-slice, full 32x32 tile.
// ---------------------------------------------------------------------------
__global__ __launch_bounds__(128) void attn_partial_kernel(
    const float* __restrict__ qkv, float* __restrict__ part)
{
  const int lane = threadIdx.x & 31;
  const int wave = threadIdx.x >> 5;
  const int l  = lane & 15;
  const int lh = lane >> 4;
  const int bh = blockIdx.y;
  const int b  = bh / NHEADS, h = bh % NHEADS;
  const int slice = blockIdx.x * 4 + wave;
  const int klen  = HWN / KSLICES;             // 1024
  const int kbeg  = slice * klen;

  const float* qb = qkv + ((size_t)b * QCC + h * HD) * HWN;
  const float* kb = qkv + ((size_t)b * QCC + 2 * CC + h * HD) * HWN;
  unsigned off = (unsigned)l * HWN + (unsigned)(kbeg + 2 * lh);

  v8f a00 = {}, a01 = {}, a10 = {}, a11 = {};
  for (int kk = 0; kk < klen; kk += 4) {
    v2f qa0 = *(const v2f*)(qb + off);
    v2f qa1 = *(const v2f*)(qb + off + 16u * HWN);
    v2f kv0 = *(const v2f*)(kb + off);
    v2f kv1 = *(const v2f*)(kb + off + 16u * HWN);
    a00 = wmma4(qa0, kv0, a00);
    a01 = wmma4(qa0, kv1, a01);
    a10 = wmma4(qa1, kv0, a10);
    a11 = wmma4(qa1, kv1, a11);
    off += 4u;
  }

  float* pp = part + ((size_t)slice * (BB * NHEADS) + bh) * (HD * HD);
#pragma unroll
  for (int r = 0; r < 8; ++r) {
    int row0 = r + 8 * lh;
    pp[(row0)      * HD + l]      = a00[r];
    pp[(row0)      * HD + 16 + l] = a01[r];
    pp[(16 + row0) * HD + l]      = a10[r];
    pp[(16 + row0) * HD + 16 + l] = a11[r];
  }
}

// ---------------------------------------------------------------------------
// Kernel 5: reduce K-slices, fold in separable norms + temperature, softmax.
// grid: B*HEADS blocks, 32 threads (thread = output row c).
// ---------------------------------------------------------------------------
__global__ __launch_bounds__(32) void attn_softmax_kernel(
    const float* __restrict__ part, const float* __restrict__ invn,
    const float* __restrict__ temp, float* __restrict__ attn)
{
  int bh = blockIdx.x;
  int b = bh / NHEADS, h = bh % NHEADS;
  int c = threadIdx.x;
  float iq = invn[b * CC + h * HD + c];
  float tp = temp[h];

  float vals[HD];
#pragma unroll
  for (int d = 0; d < HD; ++d) {
    float s = 0.f;
    for (int sl = 0; sl < KSLICES; ++sl)
      s += part[(((size_t)sl * (BB * NHEADS) + bh) * HD + c) * HD + d];
    float ik = invn[BB * CC + b * CC + h * HD + d];
    vals[d] = s * iq * ik * tp;
  }
  float mx = vals[0];
#pragma unroll
  for (int d = 1; d < HD; ++d) mx = fmaxf(mx, vals[d]);
  float sum = 0.f;
#pragma unroll
  for (int d = 0; d < HD; ++d) { vals[d] = __expf(vals[d] - mx); sum += vals[d]; }
  float inv = 1.0f / sum;
  float* ar = attn + ((size_t)bh * HD + c) * HD;
#pragma unroll
  for (int d = 0; d < HD; ++d) ar[d] = vals[d] * inv;
}

// ---------------------------------------------------------------------------
// Kernel 6: out[c][n] = (attn @ v)[c][n] * sigmoid(gate[c][n]) via WMMA.
// grid: (HW/64, B*HEADS), block 128; wave -> one 16-col n-tile, 2 M-tiles.
// ---------------------------------------------------------------------------
__global__ __launch_bounds__(128) void attnv_gate_kernel(
    const float* __restrict__ attn, const float* __restrict__ qkv,
    float* __restrict__ outC)
{
  const int lane = threadIdx.x & 31;
  const int wave = threadIdx.x >> 5;
  const int l  = lane & 15;
  const int lh = lane >> 4;
  const int bh = blockIdx.y;
  const int b  = bh / NHEADS, h = bh % NHEADS;
  const int n  = blockIdx.x * 64 + wave * 16 + l;

  const float* A  = attn + (size_t)bh * HD * HD;
  const float* vb = qkv + ((size_t)b * QCC + 3 * CC + h * HD) * HWN;
  const float* gb = qkv + ((size_t)b * QCC + 1 * CC + h * HD) * HWN;

  unsigned aoff = (unsigned)l * HD + 2u * (unsigned)lh;
  unsigned voff = (unsigned)(2 * lh) * HWN + (unsigned)n;

  v8f acc0 = {}, acc1 = {};
#pragma unroll
  for (int kk = 0; kk < HD; kk += 4) {
    v2f a0 = *(const v2f*)(A + aoff);
    v2f a1 = *(const v2f*)(A + aoff + 16u * HD);
    v2f bv;
    bv.x = vb[voff];
    bv.y = vb[voff + HWN];
    acc0 = wmma4(a0, bv, acc0);
    acc1 = wmma4(a1, bv, acc1);
    aoff += 4u;
    voff += 4u * HWN;
  }

  float* ob = outC + ((size_t)b * CC + h * HD) * HWN;
  unsigned goff = (unsigned)(8 * lh) * HWN + (unsigned)n;
#pragma unroll
  for (int r = 0; r < 8; ++r) {
    float g0 = gb[goff + (unsigned)r * HWN];
    float g1 = gb[goff + (unsigned)(16 + r) * HWN];
    ob[goff + (unsigned)r * HWN]        = acc0[r] * (1.0f / (1.0f + __expf(-g0)));
    ob[goff + (unsigned)(16 + r) * HWN] = acc1[r] * (1.0f / (1.0f + __expf(-g1)));
  }
}

// ---------------------------------------------------------------------------
extern "C" void kernel_launch(void* const* d_in, const int* in_sizes, int n_in,
                              void* d_out, int out_size, void* d_ws, size_t ws_size,
                              hipStream_t stream)
{
  const float* x      = (const float*)d_in[0];
  const float* qkv_w  = (const float*)d_in[1];
  const float* qkv_b  = (const float*)d_in[2];
  const float* dw_w   = (const float*)d_in[3];
  const float* dw_b   = (const float*)d_in[4];
  const float* temp   = (const float*)d_in[5];
  const float* proj_w = (const float*)d_in[6];
  const float* proj_b = (const float*)d_in[7];
  float* out = (float*)d_out;

  float* ws = (float*)d_ws;
  const size_t qkvN = (size_t)BB * QCC * HWN;          // 100,663,296 floats
  float* qkv0 = ws;                                    // pre-DW conv (reused as gated)
  float* qkv1 = ws + qkvN;                             // post-DW conv
  float* invn = qkv1 + qkvN;                           // 2*B*C
  float* part = invn + 2 * BB * CC;                    // KSLICES*B*HEADS*HD*HD
  float* attn = part + (size_t)KSLICES * BB * NHEADS * HD * HD;
  float* gated = qkv0;                                 // alias: qkv0 dead after DW

  // 1. QKV 1x1 conv (fp32 WMMA GEMM) + bias       M = QCC, K = CC
  gemm_wmma_kernel<<<dim3(HWN / 64, QCC / 64, BB), 128, 0, stream>>>(
      qkv_w, x, qkv_b, qkv0);
  // 2. depthwise 3x3 + bias
  dwconv_kernel<<<(unsigned)((qkvN + 255) / 256), 256, 0, stream>>>(
      qkv0, dw_w, dw_b, qkv1);
  // 3. inverse L2 norms of q/k rows
  rownorm_kernel<<<2 * BB * CC, 256, 0, stream>>>(qkv1, invn);
  // 4. split-K raw q.k^T (WMMA)
  attn_partial_kernel<<<dim3(KSLICES / 4, BB * NHEADS), 128, 0, stream>>>(
      qkv1, part);
  // 5. reduce + scale + softmax
  attn_softmax_kernel<<<BB * NHEADS, 32, 0, stream>>>(part, invn, temp, attn);
  // 6. attn @ v (WMMA) fused with sigmoid gate
  attnv_gate_kernel<<<dim3(HWN / 64, BB * NHEADS), 128, 0, stream>>>(
      attn, qkv1, gated);
  // 7. output projection (fp32 WMMA GEMM) + bias -> d_out   M = CC, K = CC
  gemm_wmma_kernel<<<dim3(HWN / 64, CC / 64, BB), 128, 0, stream>>>(
      proj_w, gated, proj_b, out);
}